// VectorQuantizer_7988639171036
// MI455X (gfx1250) — compile-verified
//
#include <hip/hip_runtime.h>
#include <hip/hip_bf16.h>

typedef __attribute__((ext_vector_type(2))) float v2f;
typedef __attribute__((ext_vector_type(8))) float v8f;

constexpr int VQ_TOK = 32768;   // B*T
constexpr int VQ_D   = 32;
constexpr int VQ_N   = 8192;

// workspace layout (in floats)
constexpr int WS_XN   = 0;                     // normalized tokens [TOK][D]
constexpr int WS_EN   = WS_XN + VQ_TOK * VQ_D; // normalized codebook [N][D]
constexpr int WS_BIAS = WS_EN + VQ_N * VQ_D;   // -0.5*||e_n||^2 [N]
constexpr int WS_IDX  = WS_BIAS + VQ_N;        // int32 argmin indices [TOK]
constexpr int WS_LOSS = WS_IDX + VQ_TOK;       // 1-float loss accumulator

// ---------------------------------------------------------------------------
// Row L2-normalize: one wave32 per row (D == warpSize == 32 on gfx1250).
// Optionally emits bias[row] = -0.5*||row_n||^2 for the codebook.
__global__ void vq_normalize_rows(const float* __restrict__ in,
                                  float* __restrict__ out,
                                  float* __restrict__ bias, int nrows) {
    int gid  = blockIdx.x * blockDim.x + threadIdx.x;
    int row  = gid >> 5;
    int lane = gid & 31;
    if (row >= nrows) return;
    float v = in[row * VQ_D + lane];
    float s = v * v;
    #pragma unroll
    for (int off = 16; off > 0; off >>= 1)
        s += __shfl_xor(s, off, 32);
    float scale = rsqrtf(s + 1e-12f);
    out[row * VQ_D + lane] = v * scale;
    if (bias != nullptr && lane == 0)
        bias[row] = -0.5f * s * scale * scale;   // -0.5*||e_n||^2 = -0.5*s/(s+eps)
}

// ---------------------------------------------------------------------------
// Fused matmul + argmax: each wave owns 16 tokens, scans all N columns in
// 16-wide tiles using V_WMMA_F32_16X16X4_F32 chains (K=32 -> 8 wmma per tile).
// C is preloaded with -0.5*||e_n||^2 so D = x.e - 0.5||e||^2; argmax == ref argmin.
__global__ void __launch_bounds__(256)
vq_argmax_wmma(const float* __restrict__ xn, const float* __restrict__ en,
               const float* __restrict__ bias, int* __restrict__ idx_ws,
               float* __restrict__ out_idx_f) {
    const int lane     = threadIdx.x & 31;
    const int wave     = threadIdx.x >> 5;
    const int tileBase = blockIdx.x * 128 + wave * 16;   // 8 waves * 16 tokens
    const int ml       = lane & 15;
    const int khalf    = (lane >> 4) * 2;                // K-offset: 0 or 2

    // A tile (16 tokens x K=32), ISA layout: one b64 per K-chunk per lane.
    v2f a[8];
    const float* arow = xn + (tileBase + ml) * VQ_D;
    #pragma unroll
    for (int k = 0; k < 8; ++k)
        a[k] = *(const v2f*)(arow + 4 * k + khalf);

    float best[8];
    int   bidx[8];
    #pragma unroll
    for (int r = 0; r < 8; ++r) { best[r] = -3.0e38f; bidx[r] = 0; }

    for (int nt = 0; nt < VQ_N / 16; ++nt) {
        const int col = nt * 16 + ml;               // this lane's column
        const float bv = bias[col];
        v8f c;
        #pragma unroll
        for (int r = 0; r < 8; ++r) c[r] = bv;      // C = -0.5*||e_col||^2

        v2f b[8];
        const float* brow = en + col * VQ_D;
        #pragma unroll
        for (int k = 0; k < 8; ++k)
            b[k] = *(const v2f*)(brow + 4 * k + khalf);

        #pragma unroll
        for (int k = 0; k < 8; ++k)
            c = __builtin_amdgcn_wmma_f32_16x16x4_f32(
                    false, a[k], false, b[k], (short)0, c, false, false);

        // branchless running argmax (keeps EXEC all-ones around WMMA)
        #pragma unroll
        for (int r = 0; r < 8; ++r) {
            bool p  = c[r] > best[r];
            bidx[r] = p ? col  : bidx[r];
            best[r] = p ? c[r] : best[r];
        }
    }

    // Cross-lane argmax within each 16-lane half (lanes share the same 8 rows).
    #pragma unroll
    for (int r = 0; r < 8; ++r) {
        #pragma unroll
        for (int off = 8; off > 0; off >>= 1) {
            float ob = __shfl_xor(best[r], off, 32);
            int   oi = __shfl_xor(bidx[r], off, 32);
            bool p = (ob > best[r]) | ((ob == best[r]) & (oi < bidx[r]));
            best[r] = p ? ob : best[r];
            bidx[r] = p ? oi : bidx[r];
        }
    }

    if (ml == 0) {  // lane 0 -> rows 0..7, lane 16 -> rows 8..15
        int mbase = (lane >> 4) * 8;
        #pragma unroll
        for (int r = 0; r < 8; ++r) {
            int token = tileBase + mbase + r;
            idx_ws[token]    = bidx[r];
            out_idx_f[token] = (float)bidx[r];
        }
    }
}

// ---------------------------------------------------------------------------
// Gather z_q = e_n[idx] and accumulate sum((z_q - x_n)^2) -> loss accumulator.
__global__ void vq_gather(const float* __restrict__ xn, const float* __restrict__ en,
                          const int* __restrict__ idx_ws, float* __restrict__ zq,
                          float* __restrict__ loss_accum) {
    int gid   = blockIdx.x * blockDim.x + threadIdx.x;  // one thread per element
    int token = gid >> 5;
    int d     = gid & 31;
    int idx   = idx_ws[token];
    float v   = en[idx * VQ_D + d];
    zq[gid]   = v;
    float diff = v - xn[gid];
    float s = diff * diff;
    #pragma unroll
    for (int off = 16; off > 0; off >>= 1)
        s += __shfl_xor(s, off, 32);
    if (d == 0) atomicAdd(loss_accum, s);
}

__global__ void vq_init(float* __restrict__ loss_accum) {
    if (threadIdx.x == 0 && blockIdx.x == 0) *loss_accum = 0.0f;
}

__global__ void vq_finalize(const float* __restrict__ loss_accum,
                            float* __restrict__ out_losses) {
    if (threadIdx.x == 0 && blockIdx.x == 0) {
        float m = *loss_accum * (1.0f / (float)(VQ_TOK * VQ_D));
        out_losses[0] = m;   // q_latent_loss (numerically identical)
        out_losses[1] = m;   // e_latent_loss
    }
}

// ---------------------------------------------------------------------------
extern "C" void kernel_launch(void* const* d_in, const int* in_sizes, int n_in,
                              void* d_out, int out_size, void* d_ws, size_t ws_size,
                              hipStream_t stream) {
    const float* hs  = (const float*)d_in[0];   // hidden_states [32,1024,32]
    const float* emb = (const float*)d_in[1];   // embedding [8192,32]

    float* ws   = (float*)d_ws;
    float* xn   = ws + WS_XN;
    float* en   = ws + WS_EN;
    float* bias = ws + WS_BIAS;
    int*   idxw = (int*)(ws + WS_IDX);
    float* lacc = ws + WS_LOSS;

    float* out     = (float*)d_out;
    float* zq      = out;                        // [TOK*D]
    float* oidx    = out + VQ_TOK * VQ_D;        // [TOK] indices as float
    float* olosses = oidx + VQ_TOK;              // [2]

    vq_init<<<1, 32, 0, stream>>>(lacc);
    vq_normalize_rows<<<(VQ_TOK * 32) / 256, 256, 0, stream>>>(hs, xn, nullptr, VQ_TOK);
    vq_normalize_rows<<<(VQ_N * 32) / 256, 256, 0, stream>>>(emb, en, bias, VQ_N);
    vq_argmax_wmma<<<VQ_TOK / 128, 256, 0, stream>>>(xn, en, bias, idxw, oidx);
    vq_gather<<<(VQ_TOK * VQ_D) / 256, 256, 0, stream>>>(xn, en, idxw, zq, lacc);
    vq_finalize<<<1, 32, 0, stream>>>(lacc, olosses);
}